// OrthogonalLayer1D_74105365725413
// MI455X (gfx1250) — compile-verified
//
#include <hip/hip_runtime.h>

// CDNA5 / gfx1250 Gram-Schmidt over [K=8, B, D=512] fp32.
// Per wave (2 samples, X_pair = 16x512):
//   phase 0: async global->LDS (ASYNCcnt) of X_pair              [fallback: b128+ds]
//   phase 1: G = X*X^T via 128x V_WMMA_F32_16X16X4_F32 (B frag == A frag)
//   phase 3: 8x8 GS coefficient recursion in Gram space (lanes 0/1)
//   phase 4: Q = C*X via 128x WMMA, Q written back into sX in place
//   phase 5: async LDS->global (ASYNCcnt) of Q                   [fallback: b32 stores]
// Memory-bound: 1 GB traffic -> ~43 us at 23.3 TB/s HBM.

typedef __attribute__((ext_vector_type(2))) float v2f;
typedef __attribute__((ext_vector_type(4))) float v4f;
typedef __attribute__((ext_vector_type(8))) float v8f;
typedef __attribute__((ext_vector_type(4))) int   v4i;

#define DDIM   512
#define XPITCH 516   // 512 + 4 pad: A-frag ds_load_b64 hits 64 distinct banks
#define CPITCH 18    // 16 + 2 pad: 8B alignment + bank spread

#if __has_builtin(__builtin_amdgcn_global_load_async_to_lds_b128) && \
    __has_builtin(__builtin_amdgcn_global_store_async_from_lds_b128)
#define HAS_ASYNC_LDS 1
typedef __attribute__((address_space(1))) v4i gv4i;  // global int4 (builtin param type)
typedef __attribute__((address_space(3))) v4i lv4i;  // lds int4
#else
#define HAS_ASYNC_LDS 0
#endif

static __device__ __forceinline__ void wait_asynccnt0() {
#if __has_builtin(__builtin_amdgcn_s_wait_asynccnt)
    __builtin_amdgcn_s_wait_asynccnt(0);
#else
    asm volatile("s_wait_asynccnt 0x0" ::: "memory");
#endif
}

__global__ __launch_bounds__(32)
void gs_wmma_kernel(const float* __restrict__ x, float* __restrict__ out, int Btot) {
    __shared__ float sX[16 * XPITCH];  // 33024 B (X_pair, later reused for Q)
    __shared__ float sG[16 * 16];      //  1024 B
    __shared__ float sC[16 * CPITCH];  //  1152 B

    const int lane = threadIdx.x;      // wave32
    const int half = lane >> 4;        // 0: lanes 0-15, 1: lanes 16-31
    const int l15  = lane & 15;
    const int b0   = blockIdx.x * 2;   // two samples per wave

    // ---- Phase 0: 16 rows (2 samples x 8 model-vectors) of 512 f32 -> LDS.
    // x layout [K][B][D]; each (k,b) row is 2KB contiguous.
#if HAS_ASYNC_LDS
    #pragma unroll
    for (int r = 0; r < 16; ++r) {
        const float* src = x + ((size_t)(r & 7) * (size_t)Btot + (size_t)(b0 + (r >> 3))) * DDIM;
        #pragma unroll
        for (int cc = 0; cc < 4; ++cc) {
            const int col = cc * 128 + lane * 4;
            __builtin_amdgcn_global_load_async_to_lds_b128(
                (gv4i*)(v4i*)(src + col),
                (lv4i*)(v4i*)&sX[r * XPITCH + col], 0, 0);
        }
    }
    wait_asynccnt0();   // LDS writes visible before ds_load use
#else
    #pragma unroll 4
    for (int r = 0; r < 16; ++r) {
        const float* src = x + ((size_t)(r & 7) * (size_t)Btot + (size_t)(b0 + (r >> 3))) * DDIM;
        #pragma unroll
        for (int cc = 0; cc < 4; ++cc) {
            const int col = cc * 128 + lane * 4;
            v4f v = *(const v4f*)(src + col);
            *(v4f*)(&sX[r * XPITCH + col]) = v;
        }
    }
#endif
    __syncthreads();   // single-wave WG: hw NOP, compiler memory fence

    // ---- Phase 1: G = X * X^T  (K swept 4 at a time).
    // A frag: lane holds A[M=l15][K = k + 2*half + {0,1}] (v2f, 8B aligned).
    // B frag of X^T is register-identical to the A frag -> pass `a` twice.
    v8f acc = {0.f, 0.f, 0.f, 0.f, 0.f, 0.f, 0.f, 0.f};
    #pragma unroll 4
    for (int k = 0; k < DDIM; k += 4) {
        v2f a = *(const v2f*)(&sX[l15 * XPITCH + k + 2 * half]);
        acc = __builtin_amdgcn_wmma_f32_16x16x4_f32(
            false, a, false, a, (short)0, acc, false, false);
    }

    // ---- Phase 2: spill G to LDS (C/D layout: VGPR v -> M = v + 8*half).
    #pragma unroll
    for (int v = 0; v < 8; ++v) {
        sG[(v + 8 * half) * 16 + l15] = acc[v];
    }
    __syncthreads();

    // ---- Phase 3: per-sample GS coefficients in Gram space.
    // Zero block-diagonal C (constant trip count: 16*18 = 9*32).
    #pragma unroll
    for (int t = 0; t < 9; ++t) sC[lane + 32 * t] = 0.f;
    if (lane < 2) {
        const int base = lane * 8;     // this sample's diagonal block
        float C[8][8];
        #pragma unroll
        for (int i = 0; i < 8; ++i) {
            float g[8];
            #pragma unroll
            for (int m = 0; m < 8; ++m)
                if (m <= i) g[m] = sG[(base + m) * 16 + (base + i)];
            float r[8];
            float nrm2 = g[i];
            #pragma unroll
            for (int j = 0; j < 8; ++j) {
                if (j < i) {
                    float s = 0.f;
                    #pragma unroll
                    for (int m = 0; m < 8; ++m)
                        if (m <= j) s += C[j][m] * g[m];   // r_j = <q_j, x_i>
                    r[j] = s;
                    nrm2 -= s * s;                         // ||w||^2 = G_ii - sum r_j^2
                }
            }
            const float nrm = sqrtf(fmaxf(nrm2, 0.f));
            const float inv = 1.f / fmaxf(nrm, 1e-6f);     // reference safe_tol
            #pragma unroll
            for (int m = 0; m < 8; ++m) {
                float c = (m == i) ? 1.f : 0.f;
                #pragma unroll
                for (int j = 0; j < 8; ++j)
                    if (j < i && m <= j) c -= r[j] * C[j][m];
                C[i][m] = (m <= i) ? c * inv : 0.f;
            }
            #pragma unroll
            for (int m = 0; m < 8; ++m)
                sC[(base + i) * CPITCH + (base + m)] = C[i][m];
        }
    }
    __syncthreads();

    // ---- Phase 4: Q = C * X, 16 cols/step, 4 chained K=4 WMMAs.
    // Q block overwrites the just-consumed X columns (LDS ops are in-order
    // within a wave; later jb blocks touch disjoint columns).
    #pragma unroll 1
    for (int jb = 0; jb < 32; ++jb) {
        const int col = jb * 16 + l15;
        v8f q = {0.f, 0.f, 0.f, 0.f, 0.f, 0.f, 0.f, 0.f};
        #pragma unroll
        for (int k4 = 0; k4 < 4; ++k4) {
            const int kk = 4 * k4 + 2 * half;
            v2f a = *(const v2f*)(&sC[l15 * CPITCH + kk]);  // A = C[M=l15][kk..kk+1]
            v2f b;
            b[0] = sX[kk * XPITCH + col];                   // B[K=kk  ][N=l15]
            b[1] = sX[(kk + 1) * XPITCH + col];             // B[K=kk+1][N=l15]
            q = __builtin_amdgcn_wmma_f32_16x16x4_f32(
                false, a, false, b, (short)0, q, false, false);
        }
#if HAS_ASYNC_LDS
        #pragma unroll
        for (int v = 0; v < 8; ++v) {
            sX[(v + 8 * half) * XPITCH + col] = q[v];       // row m = v + 8*half
        }
#else
        #pragma unroll
        for (int v = 0; v < 8; ++v) {
            out[((size_t)v * (size_t)Btot + (size_t)(b0 + half)) * DDIM + col] = q[v];
        }
#endif
    }

#if HAS_ASYNC_LDS
    // ---- Phase 5: drain Q from LDS with the async store engine.
    asm volatile("s_wait_dscnt 0x0" ::: "memory");  // Q fully in LDS first
    #pragma unroll
    for (int r = 0; r < 16; ++r) {
        float* dst = out + ((size_t)(r & 7) * (size_t)Btot + (size_t)(b0 + (r >> 3))) * DDIM;
        #pragma unroll
        for (int cc = 0; cc < 4; ++cc) {
            const int col = cc * 128 + lane * 4;
            __builtin_amdgcn_global_store_async_from_lds_b128(
                (gv4i*)(v4i*)(dst + col),
                (lv4i*)(v4i*)&sX[r * XPITCH + col], 0, 0);
        }
    }
    // S_ENDPGM performs an implicit wait-idle; stores complete before exit.
#endif
}

extern "C" void kernel_launch(void* const* d_in, const int* in_sizes, int n_in,
                              void* d_out, int out_size, void* d_ws, size_t ws_size,
                              hipStream_t stream) {
    const float* x = (const float*)d_in[0];
    float* out = (float*)d_out;
    const int total = in_sizes[0];            // K * B * D
    const int Btot  = total / (8 * DDIM);     // K=8, D=512 per reference
    const int nblk  = Btot / 2;               // 2 samples per wave/block
    gs_wmma_kernel<<<dim3(nblk), dim3(32), 0, stream>>>(x, out, Btot);
}